// rbig_block_84722524881450
// MI455X (gfx1250) — compile-verified
//
#include <hip/hip_runtime.h>
#include <hip/hip_bf16.h>
#include <math.h>

// ---------------------------------------------------------------------------
// RBIG logistic-KDE block for MI455X (gfx1250, wave32).
//  - Probability-domain mixture reduction: cdf = sum W*sigmoid(-s),
//    sf = sum W*sigmoid(s), pdf = sum (W/hs)*sig*sig  -> 1 v_exp + 1 v_rcp
//    per term, no logsumexp rescaling (W is normalized, sums bounded in [0,1]).
//  - Per-dim constants packed as float4 and broadcast from LDS (ds_load_b128).
//  - log_det row-reduction done on the matrix pipe: chained
//    v_wmma_f32_16x16x4_f32 with B = ones (K=64 in 16 steps).
// Workspace: (64 + 6*65536) floats = ~1.54 MB.
// ---------------------------------------------------------------------------

#define N_DATA 1024
#define DIM 64
#define BATCH 1024
#define HS_MIN 1e-08f
#define MASK_BOUND 5e-08f
#define LOG_HS_MIN (-18.420680743952367f)      /* log(1e-8) */
#define HALF_LOG_2PI 0.9189385332046727f
#define LOG2E 1.4426950408889634f
#define LN2 0.6931471805599453f

typedef float v8f __attribute__((ext_vector_type(8)));
typedef float v2f __attribute__((ext_vector_type(2)));

// -------------------------- ndtri (Acklam) ---------------------------------
static __device__ __forceinline__ float ndtri_f(float p) {
    const float a1 = -3.969683028665376e+01f, a2 = 2.209460984245205e+02f;
    const float a3 = -2.759285104469687e+02f, a4 = 1.383577518672690e+02f;
    const float a5 = -3.066479806614716e+01f, a6 = 2.506628277459239e+00f;
    const float b1 = -5.447609879822406e+01f, b2 = 1.615858368580409e+02f;
    const float b3 = -1.556989798598866e+02f, b4 = 6.680131188771972e+01f;
    const float b5 = -1.328068155288572e+01f;
    const float c1 = -7.784894002430293e-03f, c2 = -3.223964580411365e-01f;
    const float c3 = -2.400758277161838e+00f, c4 = -2.549732539343734e+00f;
    const float c5 =  4.374664141464968e+00f, c6 =  2.938163982698783e+00f;
    const float d1 =  7.784695709041462e-03f, d2 =  3.224671290700398e-01f;
    const float d3 =  2.445134137142996e+00f, d4 =  3.754408661907416e+00f;
    const float plow = 0.02425f, phigh = 1.0f - 0.02425f;
    float xx;
    if (p < plow) {
        float q = sqrtf(-2.0f * logf(p));
        xx = (((((c1*q+c2)*q+c3)*q+c4)*q+c5)*q+c6) /
             ((((d1*q+d2)*q+d3)*q+d4)*q+1.0f);
    } else if (p <= phigh) {
        float q = p - 0.5f, r = q*q;
        xx = (((((a1*r+a2)*r+a3)*r+a4)*r+a5)*r+a6)*q /
             (((((b1*r+b2)*r+b3)*r+b4)*r+b5)*r+1.0f);
    } else {
        float q = sqrtf(-2.0f * logf(1.0f - p));
        xx = -(((((c1*q+c2)*q+c3)*q+c4)*q+c5)*q+c6) /
              ((((d1*q+d2)*q+d3)*q+d4)*q+1.0f);
    }
    return xx;
}

// ---------------- P0: per-dim logsumexp of kde_weights ---------------------
__global__ __launch_bounds__(256) void col_lse_kernel(
        const float* __restrict__ w, float* __restrict__ lse) {
    __shared__ float red[256];
    const int d = blockIdx.x, t = threadIdx.x;
    float m = -INFINITY;
    for (int n = t; n < N_DATA; n += 256) m = fmaxf(m, w[n * DIM + d]);
    red[t] = m; __syncthreads();
    for (int s = 128; s > 0; s >>= 1) {
        if (t < s) red[t] = fmaxf(red[t], red[t + s]);
        __syncthreads();
    }
    const float mm = red[0]; __syncthreads();
    float acc = 0.0f;
    for (int n = t; n < N_DATA; n += 256)
        acc += exp2f((w[n * DIM + d] - mm) * LOG2E);
    red[t] = acc; __syncthreads();
    for (int s = 128; s > 0; s >>= 1) {
        if (t < s) red[t] += red[t + s];
        __syncthreads();
    }
    if (t == 0) lse[d] = mm + logf(red[0]);
}

// ---------------- P1: transposed packed per-point constants ----------------
// packT[d][n] = { 1/hs, dp/hs, W, W/hs_c },  dpT[d][n] = dp
__global__ __launch_bounds__(256) void precompute_kernel(
        const float* __restrict__ dp0, const float* __restrict__ delta,
        const float* __restrict__ log_hs, const float* __restrict__ kw,
        const float* __restrict__ lse,
        float4* __restrict__ packT, float* __restrict__ dpT) {
    const int idx = blockIdx.x * 256 + threadIdx.x;   // n*64 + d
    const int n = idx >> 6, d = idx & 63;
    const float lh = log_hs[idx];
    const float hs = fmaxf(exp2f(lh * LOG2E), HS_MIN);
    const float hinv = __builtin_amdgcn_rcpf(hs);
    const float dp = dp0[idx] + delta[idx];
    const float W = exp2f((kw[idx] - lse[d]) * LOG2E);
    const float lhc = fmaxf(lh, LOG_HS_MIN);
    const float whs = W * exp2f(-lhc * LOG2E);
    const int tidx = d * N_DATA + n;
    packT[tidx] = make_float4(hinv, dp * hinv, W, whs);
    dpT[tidx] = dp;
}

// ---------------- B: fused mixture reduction + epilogue --------------------
// Queries r in [0,1024): x rows (writes inverse_l + log_det term);
// r in [1024,2048): dp rows (writes cur_data_batch).
__global__ __launch_bounds__(256) void kde_main_kernel(
        const float* __restrict__ x, const float4* __restrict__ packT,
        const float* __restrict__ dpT,
        float* __restrict__ out_inv, float* __restrict__ out_cur,
        float* __restrict__ ldterm) {
    __shared__ float4 col[N_DATA];                    // 16 KB
    const int d = blockIdx.x;
    const int t = threadIdx.x;
    const float4* src = packT + d * N_DATA;
    for (int i = t; i < N_DATA; i += 256) col[i] = src[i];
    __syncthreads();

    const int r = blockIdx.y * 256 + t;               // 0..2047
    const float xv = (r < BATCH) ? x[r * DIM + d]
                                 : dpT[d * N_DATA + (r - BATCH)];

    float cdf = 0.0f, sf = 0.0f, pdf = 0.0f;
#pragma unroll 4
    for (int n = 0; n < N_DATA; ++n) {
        const float4 c = col[n];                      // ds_load_b128 broadcast
        float s = fmaf(-xv, c.x, c.y);                // (dp - x)/hs
        s = fminf(fmaxf(s, -87.0f), 87.0f);
        const float es = exp2f(s * LOG2E);            // v_exp_f32
        const float tt = __builtin_amdgcn_rcpf(1.0f + es);  // sigmoid(-s)
        const float uu = es * tt;                     // sigmoid(s)
        cdf = fmaf(c.z, tt, cdf);
        sf  = fmaf(c.z, uu, sf);
        pdf = fmaf(c.w, uu * tt, pdf);
    }

    const float log_cdf = logf(cdf);
    const float log_sf  = logf(sf);
    const float log_pdf = logf(pdf);

    const bool mid = (cdf > MASK_BOUND) && (cdf < 1.0f - MASK_BOUND);
    const bool mr  = (cdf >= 1.0f - MASK_BOUND);
    const bool ml  = (cdf <= MASK_BOUND);

    float inv = ndtri_f(mid ? cdf : 0.5f);            // ndtri(0.5) == 0
    const float sr = mr ? sqrtf(fmaxf(-2.0f * log_sf, 0.0f)) : 0.0f;
    const float sl = ml ? sqrtf(fmaxf(-2.0f * log_cdf, 0.0f)) : 0.0f;
    inv = inv + sr - sl;

    if (r < BATCH) {
        float log_g = 0.0f;
        if (mid) log_g = fmaf(-0.5f * inv, inv, -HALF_LOG_2PI);
        if (ml)  log_g = logf(sl) - log_cdf;          // masks mutually exclusive
        if (mr)  log_g = logf(sr) - log_sf;
        out_inv[r * DIM + d] = inv;
        ldterm[r * DIM + d] = log_pdf - log_g;
    } else {
        out_cur[(r - BATCH) * DIM + d] = inv;
    }
}

// ---------------- C: log_det row-sum on the matrix pipe --------------------
// Each wave32 reduces 16 rows x 64 dims via 16 chained
// v_wmma_f32_16x16x4_f32 with B = ones; rowsum(m) lands in D[m][0].
__global__ __launch_bounds__(256) void rowsum_wmma_kernel(
        const float* __restrict__ ldterm, const float* __restrict__ ld_in,
        float* __restrict__ out) {
    const int wave = threadIdx.x >> 5;                // 0..7
    const int lane = threadIdx.x & 31;
    const int rowBase = blockIdx.x * 128 + wave * 16;
    const int m = lane & 15;                          // A-matrix row
    const int hi = lane >> 4;                         // K half select

    v8f c = {};                                       // accumulator = 0
    v2f bones; bones[0] = 1.0f; bones[1] = 1.0f;      // B = ones (4x16)
    const float* rowp = ldterm + (rowBase + m) * DIM;

#pragma unroll
    for (int k = 0; k < 16; ++k) {
        v2f a;
        const int colk = 4 * k + 2 * hi;              // A 16x4 f32 layout:
        a[0] = rowp[colk];                            //  VGPR0: K = 2*hi
        a[1] = rowp[colk + 1];                        //  VGPR1: K = 2*hi + 1
        c = __builtin_amdgcn_wmma_f32_16x16x4_f32(
                false, a, false, bones, (short)0, c, false, false);
    }
    // C/D layout: lane 0 holds D[j][0] for j=0..7 ; lane 16 holds D[8+j][0]
    if (lane == 0) {
        for (int j = 0; j < 8; ++j)
            out[rowBase + j] = ld_in[rowBase + j] + c[j];
    }
    if (lane == 16) {
        for (int j = 0; j < 8; ++j)
            out[rowBase + 8 + j] = ld_in[rowBase + 8 + j] + c[j];
    }
}

// ---------------------------------------------------------------------------
extern "C" void kernel_launch(void* const* d_in, const int* in_sizes, int n_in,
                              void* d_out, int out_size, void* d_ws, size_t ws_size,
                              hipStream_t stream) {
    const float* x       = (const float*)d_in[0];   // [1024,64]
    const float* log_det = (const float*)d_in[1];   // [1024]
    const float* dp0     = (const float*)d_in[2];   // [1024,64]
    const float* delta   = (const float*)d_in[3];   // [1024,64]
    const float* log_hs  = (const float*)d_in[4];   // [1024,64]
    const float* kw      = (const float*)d_in[5];   // [1024,64]

    float* out = (float*)d_out;
    float* out_inv    = out;                         // [1024,64]
    float* out_logdet = out + BATCH * DIM;           // [1024]
    float* out_cur    = out + BATCH * DIM + BATCH;   // [1024,64]

    // Workspace layout (floats): lse[64] | packT(float4)[64*1024] |
    //                            dpT[64*1024] | ldterm[1024*64]
    float* ws = (float*)d_ws;
    float*  lse    = ws;                                    // 64
    float4* packT  = (float4*)(ws + 64);                    // 256B offset, 16B aligned
    float*  dpT    = ws + 64 + 4 * DIM * N_DATA;
    float*  ldterm = ws + 64 + 5 * DIM * N_DATA;

    col_lse_kernel<<<DIM, 256, 0, stream>>>(kw, lse);
    precompute_kernel<<<(N_DATA * DIM) / 256, 256, 0, stream>>>(
        dp0, delta, log_hs, kw, lse, packT, dpT);
    kde_main_kernel<<<dim3(DIM, (BATCH + N_DATA) / 256), 256, 0, stream>>>(
        x, packT, dpT, out_inv, out_cur, ldterm);
    rowsum_wmma_kernel<<<BATCH / 128, 256, 0, stream>>>(
        ldterm, log_det, out_logdet);
}